// StablelmAttention_1606317768925
// MI455X (gfx1250) — compile-verified
//
#include <hip/hip_runtime.h>

// ---------------- problem constants ----------------
#define BB    32
#define HH    2560
#define NH    32
#define NKV   8
#define DD    80
#define RR    20
#define LL    4096
#define QKVN  (NH*DD + 2*NKV*DD)   // 3840
#define QSZ   (NH*DD)              // 2560
#define KVSZ  (NKV*DD)             // 640
#define SCALE 0.11180339887498949f // 80^-0.5
#define KSPLIT 4

typedef __attribute__((ext_vector_type(2))) float v2f;
typedef __attribute__((ext_vector_type(8))) float v8f;

// ============================================================
// 32xN = (32xK) * (KxN) fp32 GEMM using V_WMMA_F32_16X16X4_F32.
// One wave per 32x16 tile; split-K over blockIdx.y into partial
// buffers (summed by reduce_partials for determinism).
// K and N are template constants so all in-window addresses fold
// to immediate IOFFSETs (no v_mul_u64 per load); K-loop unrolled
// x8 so loads batch into clauses and overlap with WMMA issue.
// A-layout (16x4): lane l -> M=l%16, K=j+2*(l/16)  (j = vgpr 0/1)
// B-layout (4x16): lane l -> N=l%16, K=j+2*(l/16)
// C-layout: vgpr r -> M=r+8*(l/16), N=l%16
// ============================================================
template<int K, int N>
__global__ __launch_bounds__(32)
void gemm32_wmma(const float* __restrict__ A,   // 32 x K, row major
                 const float* __restrict__ W,   // K x N,  row major
                 float* __restrict__ Cpart)     // KSPLIT x (32 x N)
{
    constexpr int KC = K / KSPLIT;
    const int n0   = blockIdx.x * 16;
    const int kc   = blockIdx.y;
    const int l    = threadIdx.x;
    const int mrow = l & 15;
    const int koff = (l >> 4) << 1;       // 0 or 2
    const int ncol = n0 + mrow;
    const int k0   = kc * KC;

    // per-lane streaming pointers (constexpr strides -> immediate offsets)
    const float* aptr0 = A + (size_t)mrow * K        + k0 + koff;
    const float* aptr1 = A + (size_t)(mrow + 16) * K + k0 + koff;
    const float* wptr  = W + (size_t)(k0 + koff) * N + ncol;

    v8f c0 = {};
    v8f c1 = {};

#pragma unroll 8
    for (int kk = 0; kk < KC; kk += 4) {
        v2f a0, a1, b;
        a0.x = aptr0[0];
        a0.y = aptr0[1];
        a1.x = aptr1[0];
        a1.y = aptr1[1];
        b.x  = wptr[0];
        b.y  = wptr[N];
        aptr0 += 4;
        aptr1 += 4;
        wptr  += 4 * (size_t)N;
        c0 = __builtin_amdgcn_wmma_f32_16x16x4_f32(false, a0, false, b, (short)0, c0, false, false);
        c1 = __builtin_amdgcn_wmma_f32_16x16x4_f32(false, a1, false, b, (short)0, c1, false, false);
    }

    const int mhi = (l >> 4) << 3;        // 0 or 8
    float* cptr = Cpart + (size_t)kc * 32 * N + (size_t)mhi * N + ncol;
#pragma unroll
    for (int r = 0; r < 8; ++r) {
        cptr[(size_t)r * N]        = c0[r];
        cptr[(size_t)(16 + r) * N] = c1[r];
    }
}

// deterministic fixed-order sum of split-K partials
__global__ void reduce_partials(const float* __restrict__ part,
                                float* __restrict__ out, int n, int nchunks)
{
    int i = blockIdx.x * blockDim.x + threadIdx.x;
    if (i < n) {
        float s = 0.f;
        for (int c = 0; c < nchunks; ++c) s += part[(size_t)c * n + i];
        out[i] = s;
    }
}

// ============================================================
// RoPE applied in-place on the qkv buffer (q heads + k heads).
// one wave per head; lanes 0..9 each handle a rotary pair.
// ============================================================
__global__ __launch_bounds__(32)
void rope_kernel(float* __restrict__ qkv, const int* __restrict__ positions)
{
    const int idx = blockIdx.x;
    const int b   = idx / (NH + NKV);
    const int h   = idx % (NH + NKV);
    float* x = (h < NH) ? (qkv + (size_t)b * QKVN + h * DD)
                        : (qkv + (size_t)b * QKVN + QSZ + (h - NH) * DD);
    const int i = threadIdx.x;
    if (i < RR / 2) {
        // inv_freq = BASE^(-2i/R) = exp(-i * (2/R) * ln(10000))
        const float inv_freq = __expf(-(float)i * 0.92103403719761840f);
        const float ang = (float)positions[b] * inv_freq;
        const float c = cosf(ang), s = sinf(ang);
        const float x1 = x[i], x2 = x[i + RR / 2];
        x[i]          = x1 * c - x2 * s;
        x[i + RR / 2] = x2 * c + x1 * s;
    }
}

// ============================================================
// GQA decode attention, one block per (b, kv-head), 320 threads.
// Pure-bandwidth kernel: stream K once (scores -> LDS), softmax
// via LDS reductions, stream V once.  LDS ~72KB (<320KB/WGP).
// ============================================================
#define NT  320
#define ST  4104        // padded score stride (L+1 = 4097)
#define LT  (LL + 1)

__global__ __launch_bounds__(NT)
void attn_kernel(const float* __restrict__ qkv,
                 const float* __restrict__ k_cache,
                 const float* __restrict__ v_cache,
                 float* __restrict__ attn_out)
{
    const int b   = blockIdx.x / NKV;
    const int kvh = blockIdx.x % NKV;
    const int t   = threadIdx.x;

    extern __shared__ float smem[];
    float* sc   = smem;                 // 4 * ST  scores
    float* qs   = sc + 4 * ST;          // 4 * 80  q heads of this group
    float* red  = qs + 4 * DD;          // 4 * NT  reduction scratch
    float* stat = red + 4 * NT;         // [0..3]=max, [4..7]=1/sum

    const float* knew = qkv + (size_t)b * QKVN + QSZ + kvh * DD;
    const float* vnew = qkv + (size_t)b * QKVN + QSZ + KVSZ + kvh * DD;

    // load the 4 query heads of this kv group into LDS (exactly 320 elems)
    {
        const int h = t / DD, d = t % DD;
        qs[t] = qkv[(size_t)b * QKVN + (size_t)(kvh * 4 + h) * DD + d];
    }
    __syncthreads();

    // ---- phase 1: scores = q . k, per-thread running max ----
    const float* kbase = k_cache + ((size_t)(b * NKV + kvh) * LL) * DD;
    float pm[4] = {-1e30f, -1e30f, -1e30f, -1e30f};

    for (int l = t; l < LT; l += NT) {
        const float* kr = (l < LL) ? (kbase + (size_t)l * DD) : knew;
        if (l + NT < LL)
            __builtin_prefetch(kbase + (size_t)(l + NT) * DD, 0, 0);
        float s0 = 0.f, s1 = 0.f, s2 = 0.f, s3 = 0.f;
#pragma unroll
        for (int d4 = 0; d4 < DD; d4 += 4) {
            const float4 k4 = *(const float4*)(kr + d4);
            s0 += qs[0 * DD + d4 + 0] * k4.x + qs[0 * DD + d4 + 1] * k4.y
                + qs[0 * DD + d4 + 2] * k4.z + qs[0 * DD + d4 + 3] * k4.w;
            s1 += qs[1 * DD + d4 + 0] * k4.x + qs[1 * DD + d4 + 1] * k4.y
                + qs[1 * DD + d4 + 2] * k4.z + qs[1 * DD + d4 + 3] * k4.w;
            s2 += qs[2 * DD + d4 + 0] * k4.x + qs[2 * DD + d4 + 1] * k4.y
                + qs[2 * DD + d4 + 2] * k4.z + qs[2 * DD + d4 + 3] * k4.w;
            s3 += qs[3 * DD + d4 + 0] * k4.x + qs[3 * DD + d4 + 1] * k4.y
                + qs[3 * DD + d4 + 2] * k4.z + qs[3 * DD + d4 + 3] * k4.w;
        }
        s0 *= SCALE; s1 *= SCALE; s2 *= SCALE; s3 *= SCALE;
        sc[0 * ST + l] = s0; sc[1 * ST + l] = s1;
        sc[2 * ST + l] = s2; sc[3 * ST + l] = s3;
        pm[0] = fmaxf(pm[0], s0); pm[1] = fmaxf(pm[1], s1);
        pm[2] = fmaxf(pm[2], s2); pm[3] = fmaxf(pm[3], s3);
    }

    // ---- block max per head ----
#pragma unroll
    for (int h = 0; h < 4; ++h) red[h * NT + t] = pm[h];
    __syncthreads();
    if (t < 4) {
        float m = -1e30f;
        for (int j = 0; j < NT; ++j) m = fmaxf(m, red[t * NT + j]);
        stat[t] = m;
    }
    __syncthreads();

    // ---- exponentiate in place, per-thread partial sums ----
    float ps[4] = {0.f, 0.f, 0.f, 0.f};
    const float m0 = stat[0], m1 = stat[1], m2 = stat[2], m3 = stat[3];
    for (int l = t; l < LT; l += NT) {
        float e0 = __expf(sc[0 * ST + l] - m0);
        float e1 = __expf(sc[1 * ST + l] - m1);
        float e2 = __expf(sc[2 * ST + l] - m2);
        float e3 = __expf(sc[3 * ST + l] - m3);
        sc[0 * ST + l] = e0; sc[1 * ST + l] = e1;
        sc[2 * ST + l] = e2; sc[3 * ST + l] = e3;
        ps[0] += e0; ps[1] += e1; ps[2] += e2; ps[3] += e3;
    }
#pragma unroll
    for (int h = 0; h < 4; ++h) red[h * NT + t] = ps[h];
    __syncthreads();
    if (t < 4) {
        float s = 0.f;
        for (int j = 0; j < NT; ++j) s += red[t * NT + j];
        stat[4 + t] = 1.0f / s;
    }
    __syncthreads();

    // ---- phase 2: out[h][d] = sum_l p[h][l] * v[l][d] ----
    {
        const int h = t / DD, d = t % DD;
        const float* vb = v_cache + ((size_t)(b * NKV + kvh) * LL) * DD + d;
        const float* ph = sc + h * ST;
        float acc = 0.f;
        for (int l = 0; l < LL; l += 16) {
#pragma unroll
            for (int j = 0; j < 16; ++j)
                acc += ph[l + j] * vb[(size_t)(l + j) * DD];
        }
        acc += ph[LL] * vnew[d];
        attn_out[(size_t)b * QSZ + (size_t)(kvh * 4 + h) * DD + d] = acc * stat[4 + h];
    }
}

// ============================================================
// host-side launch
// ============================================================
extern "C" void kernel_launch(void* const* d_in, const int* in_sizes, int n_in,
                              void* d_out, int out_size, void* d_ws, size_t ws_size,
                              hipStream_t stream)
{
    const int*   positions = (const int*)  d_in[0];
    const float* hidden    = (const float*)d_in[1];
    const float* k_cache   = (const float*)d_in[2];
    const float* v_cache   = (const float*)d_in[3];
    const float* w_qkv     = (const float*)d_in[4];
    const float* w_o       = (const float*)d_in[5];
    float*       out       = (float*)d_out;

    // workspace layout (floats)
    float* qkv  = (float*)d_ws;                 // 32*3840
    float* attn = qkv  + (size_t)BB * QKVN;     // 32*2560
    float* part = attn + (size_t)BB * QSZ;      // KSPLIT * 32 * 3840 (reused)

    // 1) qkv = hidden @ w_qkv   (split-K WMMA + deterministic reduce)
    gemm32_wmma<HH, QKVN><<<dim3(QKVN / 16, KSPLIT), 32, 0, stream>>>(
        hidden, w_qkv, part);
    reduce_partials<<<(BB * QKVN + 255) / 256, 256, 0, stream>>>(
        part, qkv, BB * QKVN, KSPLIT);

    // 2) RoPE in place on q and k slices
    rope_kernel<<<BB * (NH + NKV), 32, 0, stream>>>(qkv, positions);

    // 3) attention (memory-bound streaming kernel)
    const size_t smem = (size_t)(4 * ST + 4 * DD + 4 * NT + 8) * sizeof(float);
    attn_kernel<<<BB * NKV, NT, smem, stream>>>(qkv, k_cache, v_cache, attn);

    // 4) out = attn @ w_o
    gemm32_wmma<QSZ, HH><<<dim3(HH / 16, KSPLIT), 32, 0, stream>>>(
        attn, w_o, part);
    reduce_partials<<<(BB * HH + 255) / 256, 256, 0, stream>>>(
        part, out, BB * HH, KSPLIT);
}